// ClusteringModel_29841432773296
// MI455X (gfx1250) — compile-verified
//
#include <hip/hip_runtime.h>
#include <math.h>

// GAT layer (GATLayerImp1 eval forward) for MI455X / gfx1250, wave32.
// N=4096, H=8, F_IN=256, F_OUT=64. All f32, using V_WMMA_F32_16X16X4_F32.
//
// Pipeline:
//  K0: repack weights -> Bpack[1024][256] (proj cols 0-511 gathered from
//      [H,F_IN,F_OUT], skip cols 512-1023 = skip_w as-is) so the GEMM inner
//      loop is branch-free with aligned v2f loads.
//  K1: fused GEMM  [4096x256] x [256x1024] -> projT[512][4096] (transposed) and
//      skip[4096][512] (staged directly in d_out).
//  K1b: s_src[h][n], s_trg[h][n] dot products from projT.
//  K2: flash-attention style fused softmax(leaky_relu(s_src+s_trg^T)+mask) @ proj,
//      + skip + bias, ELU -> d_out. One 16-row tile per block, one wave per head.

typedef __attribute__((ext_vector_type(2))) float v2f;
typedef __attribute__((ext_vector_type(8))) float v8f;

#define GN 4096
#define GH 8
#define GFIN 256
#define GFOUT 64
#define GC 512          // H*F_OUT
#define GCALL 1024      // proj cols (512) + skip cols (512)

// ---------------- Kernel 0: weight repack ----------------
// Bpack[c][f]: c<512 -> proj_param[c>>6][f][c&63]; c>=512 -> skip_w[c-512][f].
__global__ __launch_bounds__(256) void gat_repack(
    const float* __restrict__ proj_param, // [8][256][64]
    const float* __restrict__ skip_w,     // [512][256]
    float* __restrict__ Bpack)            // [1024][256]
{
    const int tid = blockIdx.x * blockDim.x + threadIdx.x; // [0, 1024*256)
    const int c = tid >> 8;
    const int f = tid & (GFIN - 1);
    float v;
    if (c < GC) {
        const int h = c >> 6, o = c & 63;
        v = proj_param[(size_t)(h * GFIN + f) * GFOUT + o];
    } else {
        v = skip_w[(size_t)(c - GC) * GFIN + f];
    }
    Bpack[tid] = v;
}

// ---------------- Kernel 1: proj (transposed) + skip GEMM ----------------
// Grid: (256, 8) blocks x 256 threads (8 waves). Wave handles one 16x16 tile:
// M tile = blockIdx.x, N tile = blockIdx.y*8 + wave (64 col tiles of 16 over 1024 cols).
__global__ __launch_bounds__(256) void gat_proj_skip(
    const float* __restrict__ x,          // [4096][256]
    const float* __restrict__ Bpack,      // [1024][256]
    float* __restrict__ projT,            // [512][4096]
    float* __restrict__ skip_out)         // [4096][512] (= d_out)
{
    const int lane = threadIdx.x & 31;
    const int wave = threadIdx.x >> 5;
    const int half = lane >> 4;       // 0: lanes 0-15, 1: lanes 16-31
    const int lm   = lane & 15;
    const int n0   = blockIdx.x * 16;
    const int c0   = (blockIdx.y * 8 + wave) * 16;
    const int c    = c0 + lm;

    v8f acc = {};
    const float* xrow = x + (size_t)(n0 + lm) * GFIN + 2 * half;
    const float* brow = Bpack + (size_t)c * GFIN + 2 * half;

#pragma unroll 4
    for (int f0 = 0; f0 < GFIN; f0 += 4) {
        v2f a = *(const v2f*)(xrow + f0);   // A: M=lm, K pair {f0+2h, f0+2h+1}
        v2f b = *(const v2f*)(brow + f0);   // B: same K pair, N=c
        acc = __builtin_amdgcn_wmma_f32_16x16x4_f32(
            false, a, false, b, (short)0, acc, false, false);
    }

    // D layout: vgpr r -> row n0 + r + 8*half, col c
    if (c0 < GC) {
#pragma unroll
        for (int r = 0; r < 8; ++r) {
            const int n = n0 + r + 8 * half;
            projT[(size_t)c * GN + n] = acc[r];
        }
    } else {
#pragma unroll
        for (int r = 0; r < 8; ++r) {
            const int n = n0 + r + 8 * half;
            skip_out[(size_t)n * GC + (c - GC)] = acc[r];
        }
    }
}

// ---------------- Kernel 1b: attention source/target scores ----------------
__global__ __launch_bounds__(256) void gat_scores(
    const float* __restrict__ projT,   // [512][4096]
    const float* __restrict__ srcv,    // [8][64]
    const float* __restrict__ tgtv,    // [8][64]
    float* __restrict__ sS,            // [8][4096]
    float* __restrict__ sT)            // [8][4096]
{
    const int tid = blockIdx.x * blockDim.x + threadIdx.x; // [0, H*N)
    const int h = tid >> 12;
    const int n = tid & (GN - 1);
    float as = 0.f, at = 0.f;
#pragma unroll 8
    for (int o = 0; o < GFOUT; ++o) {
        const float p = projT[(size_t)(h * GFOUT + o) * GN + n];
        as += p * srcv[h * GFOUT + o];
        at += p * tgtv[h * GFOUT + o];
    }
    sS[tid] = as;
    sT[tid] = at;
}

// ---------------- Kernel 2: fused flash-attention + skip + ELU ----------------
// Grid: 256 blocks (16-row tiles) x 256 threads (8 waves; wave w = head w).
__global__ __launch_bounds__(256) void gat_attention(
    const float* __restrict__ projT,  // [512][4096]
    const float* __restrict__ mask,   // [4096][4096]
    const float* __restrict__ sS,     // [8][4096]
    const float* __restrict__ sT,     // [8][4096]
    const float* __restrict__ bias,   // [512]
    float* __restrict__ out)          // [4096][512]; holds skip on entry
{
    __shared__ float sbc[GH][16];     // per-wave row-broadcast scratch

    const int lane = threadIdx.x & 31;
    const int h    = threadIdx.x >> 5;   // wave index == head
    const int half = lane >> 4;
    const int i    = lane & 15;          // row within tile (A/score layout)
    const int row0 = blockIdx.x * 16;
    const int row  = row0 + i;

    const float ssrc = sS[h * GN + row];
    // per-lane streaming pointers (bumped by 16 floats per chunk)
    const float* tptr = sT + (size_t)h * GN + 2 * half;
    const float* mptr = mask + (size_t)row * GN + 2 * half;
    const float* vb0  = projT + (size_t)(h * GFOUT + 0  + i) * GN + 2 * half;
    const float* vb1  = projT + (size_t)(h * GFOUT + 16 + i) * GN + 2 * half;
    const float* vb2  = projT + (size_t)(h * GFOUT + 32 + i) * GN + 2 * half;
    const float* vb3  = projT + (size_t)(h * GFOUT + 48 + i) * GN + 2 * half;

    v8f acc[4] = {v8f{}, v8f{}, v8f{}, v8f{}};   // 16 rows x 64 cols
    float m    = -__builtin_inff();
    float ssum = 0.f;

    for (int j0 = 0; j0 < GN; j0 += 16) {
        // stream the mask ahead (global_prefetch_b8)
        __builtin_prefetch(mptr + 512, 0, 1);

        // ---- scores: leaky_relu(ssrc + strg, 0.2) + mask ----
        float sc[8];
        float cm = -__builtin_inff();
#pragma unroll
        for (int kc = 0; kc < 4; ++kc) {
            v2f t  = *(const v2f*)(tptr + 4 * kc);
            v2f mk = *(const v2f*)(mptr + 4 * kc);
            float e0 = ssrc + t.x;
            e0 = (e0 > 0.f ? e0 : 0.2f * e0) + mk.x;
            float e1 = ssrc + t.y;
            e1 = (e1 > 0.f ? e1 : 0.2f * e1) + mk.y;
            sc[2 * kc]     = e0;
            sc[2 * kc + 1] = e1;
            cm = fmaxf(cm, fmaxf(e0, e1));
        }
        // combine the two half-wave j-subsets for the full 16-wide row max
        cm = fmaxf(cm, __shfl_xor(cm, 16, 32));

        const bool  up   = cm > m;
        const float newm = fmaxf(m, cm);
        const float corr = __expf(m - newm);     // ==1 when no lane's max moved

        // ---- p = exp(score - newm), packed into A operands ----
        float psum = 0.f;
        v2f a[4];
#pragma unroll
        for (int kc = 0; kc < 4; ++kc) {
            const float p0 = __expf(sc[2 * kc] - newm);
            const float p1 = __expf(sc[2 * kc + 1] - newm);
            a[kc].x = p0;
            a[kc].y = p1;
            psum += p0 + p1;
        }
        psum += __shfl_xor(psum, 16, 32);
        ssum = ssum * corr + psum;
        m = newm;

        // ---- rescale acc only when some row's max actually increased ----
        // (wave-uniform scalar branch: EXEC stays all-ones for the WMMAs)
        if (__ballot(up)) {
            if (lane < 16) sbc[h][i] = corr;     // in-order DS within the wave
#pragma unroll
            for (int r = 0; r < 8; ++r) {
                const float cr = sbc[h][r + 8 * half];
                acc[0][r] *= cr;
                acc[1][r] *= cr;
                acc[2][r] *= cr;
                acc[3][r] *= cr;
            }
        }

        // ---- acc += P[16x16] @ V[16x64] via 16x WMMA f32 16x16x4 ----
#pragma unroll
        for (int kc = 0; kc < 4; ++kc) {
            v2f b0 = *(const v2f*)(vb0 + 4 * kc);
            v2f b1 = *(const v2f*)(vb1 + 4 * kc);
            v2f b2 = *(const v2f*)(vb2 + 4 * kc);
            v2f b3 = *(const v2f*)(vb3 + 4 * kc);
            acc[0] = __builtin_amdgcn_wmma_f32_16x16x4_f32(
                false, a[kc], false, b0, (short)0, acc[0], false, false);
            acc[1] = __builtin_amdgcn_wmma_f32_16x16x4_f32(
                false, a[kc], false, b1, (short)0, acc[1], false, false);
            acc[2] = __builtin_amdgcn_wmma_f32_16x16x4_f32(
                false, a[kc], false, b2, (short)0, acc[2], false, false);
            acc[3] = __builtin_amdgcn_wmma_f32_16x16x4_f32(
                false, a[kc], false, b3, (short)0, acc[3], false, false);
        }

        tptr += 16; mptr += 16;
        vb0 += 16; vb1 += 16; vb2 += 16; vb3 += 16;
    }

    // ---- normalize, add skip (staged in out) + bias, ELU ----
    if (lane < 16) sbc[h][i] = ssum;
#pragma unroll
    for (int r = 0; r < 8; ++r) {
        const float rinv = 1.f / sbc[h][r + 8 * half];
        const int n = row0 + r + 8 * half;
#pragma unroll
        for (int g = 0; g < 4; ++g) {
            const int c = h * GFOUT + 16 * g + i;
            float v = acc[g][r] * rinv + out[(size_t)n * GC + c] + bias[c];
            out[(size_t)n * GC + c] = (v > 0.f) ? v : expm1f(v);
        }
    }
}

// ---------------- Host-side launcher ----------------
extern "C" void kernel_launch(void* const* d_in, const int* in_sizes, int n_in,
                              void* d_out, int out_size, void* d_ws, size_t ws_size,
                              hipStream_t stream) {
    (void)in_sizes; (void)n_in; (void)out_size; (void)ws_size;

    const float* x          = (const float*)d_in[0];  // [4096][256]
    const float* mask       = (const float*)d_in[1];  // [4096][4096]
    const float* proj_param = (const float*)d_in[2];  // [8][256][64]
    const float* src        = (const float*)d_in[3];  // [8][64][1]
    const float* tgt        = (const float*)d_in[4];  // [8][64][1]
    const float* skw        = (const float*)d_in[5];  // [512][256]
    const float* bias       = (const float*)d_in[6];  // [512]
    float* out = (float*)d_out;                       // [4096][512]

    float* ws    = (float*)d_ws;
    float* projT = ws;                                // 512*4096 floats = 8 MB
    float* sS    = projT + (size_t)GC * GN;           // 8*4096
    float* sT    = sS + (size_t)GH * GN;              // 8*4096
    float* Bpack = sT + (size_t)GH * GN;              // 1024*256 floats = 1 MB

    gat_repack<<<(GCALL * GFIN) / 256, 256, 0, stream>>>(proj_param, skw, Bpack);
    dim3 g1(GN / 16, 8);
    gat_proj_skip<<<g1, 256, 0, stream>>>(x, Bpack, projT, out);
    gat_scores<<<(GH * GN) / 256, 256, 0, stream>>>(projT, src, tgt, sS, sT);
    gat_attention<<<GN / 16, 256, 0, stream>>>(projT, mask, sS, sT, bias, out);
}